// SheafDiffusionEncoder_43662637531918
// MI455X (gfx1250) — compile-verified
//
#include <hip/hip_runtime.h>
#include <math.h>

// ---------------------------------------------------------------------------
// Sheaf diffusion encoder for MI455X (gfx1250, wave32).
// Bandwidth-bound on streaming T0/T1 (2 x 409.6 MB); edge kernel streams 4KB
// transport tiles into LDS with fully coalesced float4 loads. Projection GEMM
// uses V_WMMA_F32_16X16X4_F32 (exact f32 path).
// ---------------------------------------------------------------------------

#define IN_DIM 512
#define DVEC   32

typedef float v2f __attribute__((ext_vector_type(2)));
typedef float v8f __attribute__((ext_vector_type(8)));

// ---------------------------------------------------------------- degree ----
__global__ void deg_kernel(const int* __restrict__ src, const int* __restrict__ dst,
                           float* __restrict__ deg, int E) {
    int i = blockIdx.x * blockDim.x + threadIdx.x;
    if (i < E) {
        atomicAdd(&deg[src[i]], 1.0f);
        atomicAdd(&deg[dst[i]], 1.0f);
    }
}

// ------------------------------------------------- projection via WMMA -----
// h[n, 0:32] = relu(x[n, :] @ W + b), one wave per 16-row tile, two 16x16
// accumulators covering the 32 output columns. K stepped by 4 per the
// 16x16x4 f32 WMMA shape.
__global__ void proj_wmma_kernel(const float* __restrict__ x,
                                 const float* __restrict__ W,
                                 const float* __restrict__ b,
                                 float* __restrict__ h, int N) {
    const int lane = threadIdx.x & 31;
    const int wid  = threadIdx.x >> 5;
    const int tile = blockIdx.x * 8 + wid;          // 16-row M tile
    if (tile * 16 >= N) return;                     // wave-uniform exit (EXEC stays all-1s)

    const int m0   = tile * 16;
    const int half = lane >> 4;                     // 0: K pair {k,k+1}, 1: {k+2,k+3}
    const int l    = lane & 15;

    v8f c0 = {0.f, 0.f, 0.f, 0.f, 0.f, 0.f, 0.f, 0.f};
    v8f c1 = {0.f, 0.f, 0.f, 0.f, 0.f, 0.f, 0.f, 0.f};

    const float* xrow = x + (size_t)(m0 + l) * IN_DIM;

    for (int k = 0; k < IN_DIM; k += 4) {
        const int ka = k + 2 * half;
        // A (16x4): lane(l,half) holds x[M=l, K=ka], x[M=l, K=ka+1]
        v2f a;
        a.x = xrow[ka];
        a.y = xrow[ka + 1];
        // B (4x16): lane(l,half) holds W[K=ka, N], W[K=ka+1, N]
        v2f b0, b1;
        b0.x = W[(size_t)ka * DVEC + l];
        b0.y = W[(size_t)(ka + 1) * DVEC + l];
        b1.x = W[(size_t)ka * DVEC + 16 + l];
        b1.y = W[(size_t)(ka + 1) * DVEC + 16 + l];

        c0 = __builtin_amdgcn_wmma_f32_16x16x4_f32(false, a, false, b0,
                                                   (short)0, c0, false, false);
        c1 = __builtin_amdgcn_wmma_f32_16x16x4_f32(false, a, false, b1,
                                                   (short)0, c1, false, false);
    }

    const float bias0 = b[l];
    const float bias1 = b[16 + l];
#pragma unroll
    for (int r = 0; r < 8; ++r) {
        // C/D layout: VGPR r -> M = r (lanes 0-15) / M = r+8 (lanes 16-31), N = l
        const int row = m0 + r + 8 * half;
        h[(size_t)row * DVEC + l]      = fmaxf(c0[r] + bias0, 0.f);
        h[(size_t)row * DVEC + 16 + l] = fmaxf(c1[r] + bias1, 0.f);
    }
}

// ------------------------------------------------------------ edge pass ----
// One wave per (edge, direction). g in [0, 2E): g < E is forward (uses T[g],
// reads h[src], accumulates into dst); g >= E is reverse. The 32x32 f32
// transport tile (4 KB) is loaded with coalesced float4 loads into padded LDS
// (row stride 36 floats -> conflict-free row reads), then lane d computes row
// d of T @ h_in and atomically adds c*(m - h_out_read) into h_out.
__global__ void edge_kernel(const float* __restrict__ T,
                            const float* __restrict__ rw,
                            const float* __restrict__ alpha_p,
                            const int* __restrict__ src,
                            const int* __restrict__ dst,
                            const float* __restrict__ deg,
                            const float* __restrict__ h_in,
                            float* __restrict__ h_out,
                            int E, int E2) {
    __shared__ __align__(16) float tbuf[8][32 * 36];
    __shared__ __align__(16) float hbuf[8][32];

    const int lane = threadIdx.x & 31;
    const int wid  = threadIdx.x >> 5;
    const int g    = blockIdx.x * 8 + wid;
    if (g >= E2) return;

    const bool fwd = (g < E);
    const int  e   = fwd ? g : g - E;
    const int  nin  = fwd ? src[e] : dst[e];
    const int  nout = fwd ? dst[e] : src[e];

    const float hin_l  = h_in[(size_t)nin  * DVEC + lane];
    const float hout_l = h_in[(size_t)nout * DVEC + lane];
    hbuf[wid][lane] = hin_l;

    // Cooperative, fully coalesced tile load: 256 float4 = 4 KB per wave.
    const float4* T4 = (const float4*)(T + (size_t)g * (DVEC * DVEC));
#pragma unroll
    for (int i = 0; i < 8; ++i) {
        float4 v = T4[i * 32 + lane];
        const int base = (i * 32 + lane) * 4;   // flat float index in tile
        const int row  = base >> 5;             // /32
        const int col  = base & 31;             // multiple of 4 -> 16B aligned
        *(float4*)&tbuf[wid][row * 36 + col] = v;
    }
    __builtin_amdgcn_wave_barrier();            // order LDS writes vs cross-lane reads

    const float4* trow4 = (const float4*)&tbuf[wid][lane * 36];
    const float4* hb4   = (const float4*)&hbuf[wid][0];
    float acc = 0.f;
#pragma unroll
    for (int i = 0; i < 8; ++i) {
        const float4 t = trow4[i];
        const float4 hv = hb4[i];
        acc += t.x * hv.x + t.y * hv.y + t.z * hv.z + t.w * hv.w;
    }

    const float alpha = alpha_p[0];
    const float w     = log1pf(expf(rw[g]));            // softplus
    const float c     = alpha * w / fmaxf(deg[nout], 1.0f);

    atomicAdd(&h_out[(size_t)nout * DVEC + lane], c * (acc - hout_l));
}

// ---------------------------------------------------- layernorm + relu -----
__global__ void ln_relu_kernel(const float* __restrict__ hin,
                               float* __restrict__ hout,
                               const float* __restrict__ gam,
                               const float* __restrict__ bet, int N) {
    const int lane = threadIdx.x & 31;
    const int wid  = threadIdx.x >> 5;
    const int n    = blockIdx.x * 8 + wid;
    if (n >= N) return;

    const float v = hin[(size_t)n * DVEC + lane];

    float s = v;
#pragma unroll
    for (int off = 16; off >= 1; off >>= 1) s += __shfl_xor(s, off, 32);
    const float mu = s * (1.0f / DVEC);

    const float d  = v - mu;
    float s2 = d * d;
#pragma unroll
    for (int off = 16; off >= 1; off >>= 1) s2 += __shfl_xor(s2, off, 32);
    const float var = s2 * (1.0f / DVEC);

    const float y = d * rsqrtf(var + 1e-5f) * gam[lane] + bet[lane];
    hout[(size_t)n * DVEC + lane] = fmaxf(y, 0.f);
}

// ---------------------------------------------------------------------------
extern "C" void kernel_launch(void* const* d_in, const int* in_sizes, int n_in,
                              void* d_out, int out_size, void* d_ws, size_t ws_size,
                              hipStream_t stream) {
    const float* x   = (const float*)d_in[0];
    const int*   ei  = (const int*)d_in[1];
    const float* W   = (const float*)d_in[2];
    const float* b   = (const float*)d_in[3];
    const float* al0 = (const float*)d_in[4];
    const float* T0  = (const float*)d_in[5];
    const float* rw0 = (const float*)d_in[6];
    const float* g0  = (const float*)d_in[7];
    const float* be0 = (const float*)d_in[8];
    const float* al1 = (const float*)d_in[9];
    const float* T1  = (const float*)d_in[10];
    const float* rw1 = (const float*)d_in[11];
    const float* g1  = (const float*)d_in[12];
    const float* be1 = (const float*)d_in[13];

    const int N = in_sizes[0] / IN_DIM;   // 20000
    const int E = in_sizes[1] / 2;        // 50000
    const int E2 = 2 * E;

    const int* src = ei;
    const int* dst = ei + E;

    // workspace layout: deg[N] | hA[N*32] | hB[N*32]
    float* deg = (float*)d_ws;
    float* hA  = deg + N;
    float* hB  = hA + (size_t)N * DVEC;
    (void)ws_size;

    const size_t hbytes = (size_t)N * DVEC * sizeof(float);
    float* out = (float*)d_out;

    // degree
    hipMemsetAsync(deg, 0, (size_t)N * sizeof(float), stream);
    deg_kernel<<<(E + 255) / 256, 256, 0, stream>>>(src, dst, deg, E);

    // input projection (WMMA) -> hA
    {
        const int tiles  = (N + 15) / 16;
        const int blocks = (tiles + 7) / 8;
        proj_wmma_kernel<<<blocks, 256, 0, stream>>>(x, W, b, hA, N);
    }

    // ---- layer 0 ----
    hipMemcpyAsync(hB, hA, hbytes, hipMemcpyDeviceToDevice, stream);
    edge_kernel<<<(E2 + 7) / 8, 256, 0, stream>>>(T0, rw0, al0, src, dst, deg, hA, hB, E, E2);
    ln_relu_kernel<<<(N + 7) / 8, 256, 0, stream>>>(hB, hA, g0, be0, N);

    // ---- layer 1 ----
    hipMemcpyAsync(hB, hA, hbytes, hipMemcpyDeviceToDevice, stream);
    edge_kernel<<<(E2 + 7) / 8, 256, 0, stream>>>(T1, rw1, al1, src, dst, deg, hA, hB, E, E2);
    ln_relu_kernel<<<(N + 7) / 8, 256, 0, stream>>>(hB, out, g1, be1, N);
}